// VonMisesBaseDistribution_80908593922313
// MI455X (gfx1250) — compile-verified
//
#include <hip/hip_runtime.h>
#include <math.h>

typedef __attribute__((ext_vector_type(4))) float v4f;

#define PI_F      3.14159265358979323846f
#define LOG2PI_F  1.8378770664093453f
#define PIPE      4          // async global->LDS pipeline depth
#define SEG       1024       // floats per block-row chunk (256 thr x float4)

// ---------------------------------------------------------------------------
// Kernel 1: per-column parameters (D elements, trivial cost, L2-resident)
//   k     = softplus(cu)
//   shift = pi + loc              (value - pi - loc == value - shift)
//   off   = -(log(2pi) + logI0(k))
// ---------------------------------------------------------------------------
__global__ __launch_bounds__(256) void vm_params_kernel(
    const float* __restrict__ loc,
    const float* __restrict__ cu,
    float* __restrict__ kArr,
    float* __restrict__ shiftArr,
    float* __restrict__ offArr,
    int D)
{
    int d = blockIdx.x * blockDim.x + threadIdx.x;
    if (d >= D) return;

    float c = cu[d];
    // numerically stable softplus
    float k = fmaxf(c, 0.0f) + log1pf(__expf(-fabsf(c)));

    float logI0;
    if (k < 3.75f) {
        float y = k * (1.0f / 3.75f);
        y = y * y;
        float p = 0.0045813f;
        p = fmaf(p, y, 0.0360768f);
        p = fmaf(p, y, 0.2659732f);
        p = fmaf(p, y, 1.2067492f);
        p = fmaf(p, y, 3.0899424f);
        p = fmaf(p, y, 3.5156229f);
        p = fmaf(p, y, 1.0f);
        logI0 = __logf(p);
    } else {
        float y = 3.75f / k;
        float p = 0.00392377f;
        p = fmaf(p, y, -0.01647633f);
        p = fmaf(p, y,  0.02635537f);
        p = fmaf(p, y, -0.02057706f);
        p = fmaf(p, y,  0.00916281f);
        p = fmaf(p, y, -0.00157565f);
        p = fmaf(p, y,  0.00225319f);
        p = fmaf(p, y,  0.01328592f);
        p = fmaf(p, y,  0.39894228f);
        logI0 = k - 0.5f * __logf(k) + __logf(p);
    }

    kArr[d]     = k;
    shiftArr[d] = PI_F + loc[d];
    offArr[d]   = -(LOG2PI_F + logI0);
}

// ---------------------------------------------------------------------------
// Kernel 2: main streaming pass. Each block owns a 1024-float column segment
// (256 threads x float4). Rows flow through a 4-deep LDS pipeline filled by
// CDNA5 async global->LDS B128 copies (ASYNCcnt). Each thread reads back only
// the 16 bytes it fetched itself, so no barriers are needed — just
// s_wait_asynccnt. 4 outstanding copies/wave (~2 KB) for HBM latency hiding.
// ---------------------------------------------------------------------------
__global__ __launch_bounds__(256) void vm_main_kernel(
    const float* __restrict__ value,
    const float* __restrict__ kArr,
    const float* __restrict__ shiftArr,
    const float* __restrict__ offArr,
    float* __restrict__ out,
    int S, int D)
{
    __shared__ float buf[PIPE][SEG];

    const int tid = threadIdx.x;
    const int c0  = blockIdx.x * SEG + tid * 4;

    // column params: loaded once, reused across all rows this block handles
    v4f k4  = *(const v4f*)(kArr     + c0);
    v4f sh4 = *(const v4f*)(shiftArr + c0);
    v4f of4 = *(const v4f*)(offArr   + c0);

    // LDS byte address of this thread's private 16B slot in buffer 0
    // (low 32 bits of a generic pointer to LDS == LDS byte address)
    const unsigned int ldsBase = (unsigned int)(uintptr_t)(&buf[0][tid * 4]);

    const int r0 = blockIdx.y;
    if (r0 >= S) return;
    const int rstep = gridDim.y;
    const int n = (S - 1 - r0) / rstep + 1;     // iterations for this block

    const size_t rowStride = (size_t)D;

    // ---- pipeline fill: issue up to PIPE async copies ----
    {
        const int pre = n < PIPE ? n : PIPE;
        for (int i = 0; i < pre; ++i) {
            unsigned long long g = (unsigned long long)(uintptr_t)
                (value + (size_t)(r0 + i * rstep) * rowStride + c0);
            unsigned int lds = ldsBase + (unsigned)(i & (PIPE - 1)) * (SEG * 4);
            asm volatile("global_load_async_to_lds_b128 %0, %1, off th:TH_LOAD_NT"
                         :: "v"(lds), "v"(g) : "memory");
        }
    }

    // ---- steady state ----
    for (int i = 0; i < n; ++i) {
        const int j = i + PIPE;
        if (j < n) {
            unsigned long long g = (unsigned long long)(uintptr_t)
                (value + (size_t)(r0 + j * rstep) * rowStride + c0);
            unsigned int lds = ldsBase + (unsigned)(j & (PIPE - 1)) * (SEG * 4);
            asm volatile("global_load_async_to_lds_b128 %0, %1, off th:TH_LOAD_NT"
                         :: "v"(lds), "v"(g) : "memory");
            // after issuing j, <=PIPE outstanding; draining to PIPE-1
            // guarantees buffer i has landed in LDS
            asm volatile("s_wait_asynccnt 0x3" ::: "memory");
        } else {
            // tail: nothing left to issue; drain everything (conservative,
            // touches only the last PIPE iterations, all already in flight)
            asm volatile("s_wait_asynccnt 0x0" ::: "memory");
        }

        v4f v = *(const v4f*)(&buf[i & (PIPE - 1)][tid * 4]);   // ds_load_b128

        v4f res;
        res.x = fmaf(k4.x, __cosf(v.x - sh4.x), of4.x);
        res.y = fmaf(k4.y, __cosf(v.y - sh4.y), of4.y);
        res.z = fmaf(k4.z, __cosf(v.z - sh4.z), of4.z);
        res.w = fmaf(k4.w, __cosf(v.w - sh4.w), of4.w);

        // write-once output: non-temporal b128 store, keep L2 for params
        __builtin_nontemporal_store(
            res, (v4f*)(out + (size_t)(r0 + i * rstep) * rowStride + c0));
    }
}

// ---------------------------------------------------------------------------
// Fallback for shapes whose row length is not a multiple of 1024 floats
// ---------------------------------------------------------------------------
__global__ __launch_bounds__(256) void vm_fallback_kernel(
    const float* __restrict__ value,
    const float* __restrict__ kArr,
    const float* __restrict__ shiftArr,
    const float* __restrict__ offArr,
    float* __restrict__ out,
    long long N, int D)
{
    long long i = (long long)blockIdx.x * blockDim.x + threadIdx.x;
    long long stride = (long long)gridDim.x * blockDim.x;
    for (; i < N; i += stride) {
        int d = (int)(i % D);
        out[i] = fmaf(kArr[d], __cosf(value[i] - shiftArr[d]), offArr[d]);
    }
}

extern "C" void kernel_launch(void* const* d_in, const int* in_sizes, int n_in,
                              void* d_out, int out_size, void* d_ws, size_t ws_size,
                              hipStream_t stream) {
    const float* value = (const float*)d_in[0];
    const float* loc   = (const float*)d_in[1];
    const float* cu    = (const float*)d_in[2];

    const int D = in_sizes[1];            // 8192
    const int S = in_sizes[0] / D;        // 4096

    float* kArr     = (float*)d_ws;       // 3*D floats of scratch (96 KB)
    float* shiftArr = kArr + D;
    float* offArr   = shiftArr + D;
    float* out      = (float*)d_out;

    vm_params_kernel<<<(D + 255) / 256, 256, 0, stream>>>(
        loc, cu, kArr, shiftArr, offArr, D);

    if ((D % SEG) == 0 && S > 0) {
        unsigned gy = (unsigned)(S < 256 ? S : 256);
        dim3 grid((unsigned)(D / SEG), gy);
        vm_main_kernel<<<grid, 256, 0, stream>>>(
            value, kArr, shiftArr, offArr, out, S, D);
    } else {
        long long N = (long long)S * D;
        long long want = (N + 255) / 256;
        int blocks = (int)(want < 4096 ? (want > 0 ? want : 1) : 4096);
        vm_fallback_kernel<<<blocks, 256, 0, stream>>>(
            value, kArr, shiftArr, offArr, out, N, D);
    }
}